// HeteroGNN_45689862094941
// MI455X (gfx1250) — compile-verified
//
#include <hip/hip_runtime.h>

#define N_NODES 50000
#define DIM     128
#define N_EDGES 1600000
#define OUT_DIM 2

typedef float v2f __attribute__((ext_vector_type(2)));
typedef float v8f __attribute__((ext_vector_type(8)));

// ---------------------------------------------------------------- utilities
__global__ void zero_f32(float* __restrict__ p, long n) {
  long i = (long)blockIdx.x * blockDim.x + threadIdx.x;
  long stride = (long)gridDim.x * blockDim.x;
  for (; i < n; i += stride) p[i] = 0.0f;
}

// one thread per edge: degree count (float, matches reference segment_sum of ones)
__global__ void count_edges(const int* __restrict__ dst, float* __restrict__ cnt, int ne) {
  int e = blockIdx.x * blockDim.x + threadIdx.x;
  if (e < ne) atomicAdd(&cnt[dst[e]], 1.0f);
}

// one wave32 per edge; each lane moves float4 (32*4 = 128 floats)
__global__ __launch_bounds__(256)
void scatter_feat(const float* __restrict__ x, const int* __restrict__ src,
                  const int* __restrict__ dst, float* __restrict__ sum, int ne) {
  int t = blockIdx.x * blockDim.x + threadIdx.x;
  int edge = t >> 5;
  int lane = t & 31;
  if (edge >= ne) return;
  int s = src[edge];
  int d = dst[edge];
  float4 v = ((const float4*)(x + (size_t)s * DIM))[lane];
  float* o = sum + (size_t)d * DIM + lane * 4;
  atomicAdd(o + 0, v.x);
  atomicAdd(o + 1, v.y);
  atomicAdd(o + 2, v.z);
  atomicAdd(o + 3, v.w);
}

// sum[n][d] /= max(cnt[n], 1)
__global__ void divide_mean(float* __restrict__ sum, const float* __restrict__ cnt, long n) {
  long i = (long)blockIdx.x * blockDim.x + threadIdx.x;
  if (i < n) {
    float c = cnt[i >> 7];               // i / DIM
    sum[i] = sum[i] / fmaxf(c, 1.0f);
  }
}

// ---------------------------------------------------------------- fused SAGE GEMM
// out[r][c] = act( A1[r][:] @ W1[:][c] + A2[r][:] @ W2[:][c] + bias[c] )
// One wave per 16x16 tile, K stepped by 4 using V_WMMA_F32_16X16X4_F32.
// All control flow before the WMMA is wave-uniform -> EXEC all ones.
__global__ __launch_bounds__(256)
void sage_gemm_f32(const float* __restrict__ A1, const float* __restrict__ W1,
                   const float* __restrict__ A2, const float* __restrict__ W2,
                   const float* __restrict__ bias, float* __restrict__ out,
                   int nrows, int do_relu) {
  const int tilesN = DIM / 16;                      // 8 column tiles
  int wave = (blockIdx.x * blockDim.x + threadIdx.x) >> 5;
  int lane = threadIdx.x & 31;
  int mTile = wave / tilesN;
  int nTile = wave % tilesN;
  if (mTile * 16 >= nrows) return;                  // wave-uniform

  int row0 = mTile * 16;
  int col0 = nTile * 16;
  int mrow  = lane & 15;   // A: M index held by this lane
  int khalf = lane >> 4;   // A/B: 0 -> K={0,1}, 1 -> K={2,3} within the 4-step
  int ncol  = lane & 15;   // B/C: N index held by this lane

  v8f acc = {};
  for (int k = 0; k < DIM; k += 4) {
    int kk = k + 2 * khalf;
    // A fragment (16x4, MxK): lane holds A[row][kk], A[row][kk+1]
    const float* a1p = A1 + (size_t)(row0 + mrow) * DIM + kk;
    v2f a1; a1.x = a1p[0]; a1.y = a1p[1];
    // B fragment (4x16, KxN): lane holds W[kk][col], W[kk+1][col]
    const float* b1p = W1 + (size_t)kk * DIM + col0 + ncol;
    v2f b1; b1.x = b1p[0]; b1.y = b1p[DIM];
    acc = __builtin_amdgcn_wmma_f32_16x16x4_f32(false, a1, false, b1,
                                                (short)0, acc, false, false);

    const float* a2p = A2 + (size_t)(row0 + mrow) * DIM + kk;
    v2f a2; a2.x = a2p[0]; a2.y = a2p[1];
    const float* b2p = W2 + (size_t)kk * DIM + col0 + ncol;
    v2f b2; b2.x = b2p[0]; b2.y = b2p[DIM];
    acc = __builtin_amdgcn_wmma_f32_16x16x4_f32(false, a2, false, b2,
                                                (short)0, acc, false, false);
  }

  // C/D 16x16 f32 layout: VGPR v: lane<16 -> M=v, lane>=16 -> M=v+8; N = lane&15
  int mAdd = (lane >> 4) * 8;
  for (int v = 0; v < 8; ++v) {
    int r = row0 + v + mAdd;
    int c = col0 + ncol;
    float val = acc[v] + bias[c];
    if (do_relu) val = fmaxf(val, 0.0f);
    out[(size_t)r * DIM + c] = val;
  }
}

// ---------------------------------------------------------------- final [128 -> 2]
__global__ void final_linear(const float* __restrict__ h, const float* __restrict__ W,
                             const float* __restrict__ b, float* __restrict__ out) {
  int n = blockIdx.x * blockDim.x + threadIdx.x;
  if (n >= N_NODES) return;
  const float4* hp = (const float4*)(h + (size_t)n * DIM);
  float o0 = b[0], o1 = b[1];
#pragma unroll 8
  for (int d4 = 0; d4 < DIM / 4; ++d4) {
    float4 v = hp[d4];
    int d = d4 * 4;
    o0 += v.x * W[(d + 0) * OUT_DIM + 0] + v.y * W[(d + 1) * OUT_DIM + 0] +
          v.z * W[(d + 2) * OUT_DIM + 0] + v.w * W[(d + 3) * OUT_DIM + 0];
    o1 += v.x * W[(d + 0) * OUT_DIM + 1] + v.y * W[(d + 1) * OUT_DIM + 1] +
          v.z * W[(d + 2) * OUT_DIM + 1] + v.w * W[(d + 3) * OUT_DIM + 1];
  }
  out[(size_t)n * OUT_DIM + 0] = o0;
  out[(size_t)n * OUT_DIM + 1] = o1;
}

// ---------------------------------------------------------------- launch
extern "C" void kernel_launch(void* const* d_in, const int* in_sizes, int n_in,
                              void* d_out, int out_size, void* d_ws, size_t ws_size,
                              hipStream_t stream) {
  const float* x_student = (const float*)d_in[0];
  const float* x_studies = (const float*)d_in[1];
  const int*   ei_s2st   = (const int*)d_in[2];   // [2,E] row0=src, row1=dst
  const int*   ei_st2s   = (const int*)d_in[3];
  const float* l1_st_Wl  = (const float*)d_in[4];
  const float* l1_st_b   = (const float*)d_in[5];
  const float* l1_st_Wr  = (const float*)d_in[6];
  const float* l1_s_Wl   = (const float*)d_in[7];
  const float* l1_s_b    = (const float*)d_in[8];
  const float* l1_s_Wr   = (const float*)d_in[9];
  const float* l2_st_Wl  = (const float*)d_in[10];
  const float* l2_st_b   = (const float*)d_in[11];
  const float* l2_st_Wr  = (const float*)d_in[12];
  // d_in[13..15] (l2_s_*) unused by the reference
  const float* lin_W     = (const float*)d_in[16];
  const float* lin_b     = (const float*)d_in[17];
  float* out = (float*)d_out;

  const size_t featBytes = (size_t)N_NODES * DIM * sizeof(float);
  char* ws = (char*)d_ws;
  float* sum_st = (float*)ws; ws += featBytes;   // agg into 'studies' (layer1 & layer2)
  float* sum_s  = (float*)ws; ws += featBytes;   // agg into 'student' (layer1); later h_st2
  float* h_st   = (float*)ws; ws += featBytes;
  float* h_s    = (float*)ws; ws += featBytes;
  float* cnt_st = (float*)ws; ws += (size_t)N_NODES * sizeof(float);
  float* cnt_s  = (float*)ws; ws += (size_t)N_NODES * sizeof(float);

  const long nFeat = (long)N_NODES * DIM;
  const int zeroBlocks = 4096;
  const int scatterBlocks = (N_EDGES * 32 + 255) / 256;       // one wave per edge
  const int cntBlocks = (N_EDGES + 255) / 256;
  const int meanBlocks = (int)((nFeat + 255) / 256);
  const int gemmBlocks = (N_NODES / 16) * (DIM / 16) / 8;     // 8 waves per block

  // ---- layer 1 aggregation
  zero_f32<<<zeroBlocks, 256, 0, stream>>>(sum_st, nFeat);
  zero_f32<<<zeroBlocks, 256, 0, stream>>>(sum_s, nFeat);
  zero_f32<<<(N_NODES + 255) / 256, 256, 0, stream>>>(cnt_st, N_NODES);
  zero_f32<<<(N_NODES + 255) / 256, 256, 0, stream>>>(cnt_s, N_NODES);

  count_edges<<<cntBlocks, 256, 0, stream>>>(ei_s2st + N_EDGES, cnt_st, N_EDGES);
  count_edges<<<cntBlocks, 256, 0, stream>>>(ei_st2s + N_EDGES, cnt_s, N_EDGES);

  scatter_feat<<<scatterBlocks, 256, 0, stream>>>(x_student, ei_s2st, ei_s2st + N_EDGES,
                                                  sum_st, N_EDGES);
  scatter_feat<<<scatterBlocks, 256, 0, stream>>>(x_studies, ei_st2s, ei_st2s + N_EDGES,
                                                  sum_s, N_EDGES);

  divide_mean<<<meanBlocks, 256, 0, stream>>>(sum_st, cnt_st, nFeat);
  divide_mean<<<meanBlocks, 256, 0, stream>>>(sum_s, cnt_s, nFeat);

  // ---- layer 1 SAGE GEMMs (WMMA f32), with ReLU
  sage_gemm_f32<<<gemmBlocks, 256, 0, stream>>>(sum_st, l1_st_Wl, x_studies, l1_st_Wr,
                                                l1_st_b, h_st, N_NODES, 1);
  sage_gemm_f32<<<gemmBlocks, 256, 0, stream>>>(sum_s, l1_s_Wl, x_student, l1_s_Wr,
                                                l1_s_b, h_s, N_NODES, 1);

  // ---- layer 2 aggregation over the same edge set (cnt_st still valid)
  zero_f32<<<zeroBlocks, 256, 0, stream>>>(sum_st, nFeat);
  scatter_feat<<<scatterBlocks, 256, 0, stream>>>(h_s, ei_s2st, ei_s2st + N_EDGES,
                                                  sum_st, N_EDGES);
  divide_mean<<<meanBlocks, 256, 0, stream>>>(sum_st, cnt_st, nFeat);

  // ---- layer 2 SAGE GEMM (no ReLU); write h_st2 into sum_s (now free)
  sage_gemm_f32<<<gemmBlocks, 256, 0, stream>>>(sum_st, l2_st_Wl, h_st, l2_st_Wr,
                                                l2_st_b, sum_s, N_NODES, 0);

  // ---- final projection [N,128] @ [128,2] + b
  final_linear<<<(N_NODES + 255) / 256, 256, 0, stream>>>(sum_s, lin_W, lin_b, out);
}